// QuantizationLayerBinary_55078660604231
// MI455X (gfx1250) — compile-verified
//
#include <hip/hip_runtime.h>

// Problem constants (match reference)
#define CBINS   16
#define Hd      260
#define Wd      346
#define NUM_VOX (2 * CBINS * Hd * Wd)   // 2,878,720 f32 (~11.5 MB, fits in 192MB L2)

typedef float f4 __attribute__((ext_vector_type(4)));

// ws layout (uint32 words):
//   ws[0] = deferred count (p==0 events)     [atomicAdd]
//   ws[1] = tmax bit pattern                 [atomicMax; t>0 => uint order == float order]
//   ws[2] = overflow flag                    [benign-race store of 1]
//   ws[8..8+cap) = deferred event indices
#define WS_LIST_OFF 8u

__device__ __forceinline__ void scatter_one(float x, float y, float t, float p,
                                            float* __restrict__ out) {
  // valid = (t > 0) & (t <= 1); NaN compares false -> dropped (matches jnp drop mode)
  bool valid = (t > 0.0f) && (t <= 1.0f);
  float ibin = __builtin_ceilf(t * (float)CBINS) - 1.0f;   // ceil(t*C) - 1
  float p01  = (p + 1.0f) * 0.5f;                          // {-1,1} -> {0,1}
  // All addends are integers (or .5 multiples) < 2^24 -> exact in f32,
  // matching the reference's float index arithmetic bit-for-bit.
  float idxf = x
             + (float)Wd * y
             + (float)(Wd * Hd * CBINS) * p01
             + (float)(Wd * Hd) * ibin;
  int idx = (int)idxf;
  if (valid && (unsigned)idx < (unsigned)NUM_VOX) {
    out[idx] = 1.0f;   // idempotent set: no atomic needed, all writers store 1.0
  }
}

// ---------------------------------------------------------------------------
// Kernel 1: zero the output voxel grid with 128-bit stores; reset ws header.
// ---------------------------------------------------------------------------
__global__ __launch_bounds__(256) void vox_zero_kernel(float* __restrict__ out,
                                                       unsigned n_elems,
                                                       unsigned* __restrict__ ws) {
  unsigned i = blockIdx.x * blockDim.x + threadIdx.x;
  if (i == 0) { ws[0] = 0u; ws[1] = 0u; ws[2] = 0u; }
  unsigned n4 = n_elems >> 2;
  if (i < n4) {
    ((f4*)out)[i] = (f4){0.0f, 0.0f, 0.0f, 0.0f};
  } else if (i == n4) {
    for (unsigned r = n4 << 2; r < n_elems; ++r) out[r] = 0.0f;  // tail (none here)
  }
}

// ---------------------------------------------------------------------------
// Kernel 2 (the single streaming pass): one event per lane, one nontemporal
// global_load_b128, a few VALU ops, one plain global_store_b32 for valid
// events. p==0 events (normalization candidates; empty set for the given
// input) are deferred: tmax accumulated via atomicMax on the bit pattern,
// index appended to the ws list (overflow flag if capacity exceeded).
// ---------------------------------------------------------------------------
__global__ __launch_bounds__(256) void fused_scatter_kernel(const float* __restrict__ ev,
                                                            int n,
                                                            float* __restrict__ out,
                                                            unsigned* __restrict__ ws,
                                                            unsigned cap) {
  int i = blockIdx.x * blockDim.x + threadIdx.x;
  if (i >= n) return;

  f4 e = __builtin_nontemporal_load((const f4*)ev + i);

  if (e.w != 0.0f) {
    // Normalization provably never applies when p != 0 (mask m is false).
    scatter_one(e.x, e.y, e.z, e.w, out);
  } else {
    // Rare/never path: record tmax candidate and defer the event.
    atomicMax(ws + 1, __float_as_uint(e.z));
    unsigned pos = atomicAdd(ws + 0, 1u);
    if (pos < cap) ws[WS_LIST_OFF + pos] = (unsigned)i;
    else           ws[2] = 1u;              // benign race: all writers store 1
  }
}

// ---------------------------------------------------------------------------
// Kernel 3 (cleanup, grid-stride): processes deferred p==0 events with
// t -> t/tmax. Expected count == 0 -> scalar header load + immediate exit.
// On overflow, falls back to re-scanning the whole event array (correct,
// idempotent; double-processing listed events is harmless).
// ---------------------------------------------------------------------------
__global__ __launch_bounds__(256) void deferred_kernel(const float* __restrict__ ev,
                                                       int n,
                                                       float* __restrict__ out,
                                                       const unsigned* __restrict__ ws,
                                                       unsigned cap) {
  unsigned count = ws[0];
  unsigned ovf   = ws[2];
  float    tmax  = __uint_as_float(ws[1]);
  unsigned tid    = blockIdx.x * blockDim.x + threadIdx.x;
  unsigned stride = gridDim.x * blockDim.x;

  if (!ovf) {
    unsigned m = (count < cap) ? count : cap;
    for (unsigned k = tid; k < m; k += stride) {
      unsigned j = ws[WS_LIST_OFF + k];
      f4 e = *((const f4*)ev + j);
      scatter_one(e.x, e.y, e.z / tmax, e.w, out);
    }
  } else {
    for (unsigned k = tid; k < (unsigned)n; k += stride) {
      f4 e = __builtin_nontemporal_load((const f4*)ev + k);
      if (e.w == 0.0f) scatter_one(e.x, e.y, e.z / tmax, e.w, out);
    }
  }
}

// ---------------------------------------------------------------------------
// Launch: zero-fill -> fused streaming scatter -> (expected-empty) cleanup.
// All stream-ordered; deterministic fixed work per call.
// ---------------------------------------------------------------------------
extern "C" void kernel_launch(void* const* d_in, const int* in_sizes, int n_in,
                              void* d_out, int out_size, void* d_ws, size_t ws_size,
                              hipStream_t stream) {
  const float* ev = (const float*)d_in[0];     // events, [N,4] row-major f32
  int n = in_sizes[0] / 4;                     // N events
  float* out = (float*)d_out;                  // NUM_VOX f32
  unsigned* ws = (unsigned*)d_ws;

  // Deferred-list capacity in uint32 entries (header reserves 8 words).
  unsigned cap = (ws_size >= 64) ? (unsigned)((ws_size - 64) / 4) : 0u;

  {
    unsigned n_elems = (unsigned)out_size;
    unsigned work = (n_elems >> 2) + 1;        // f4 chunks + tail thread
    unsigned blocks = (work + 255u) / 256u;
    vox_zero_kernel<<<blocks, 256, 0, stream>>>(out, n_elems, ws);
  }
  {
    int blocks = (n + 255) / 256;
    fused_scatter_kernel<<<blocks, 256, 0, stream>>>(ev, n, out, ws, cap);
  }
  {
    deferred_kernel<<<2048, 256, 0, stream>>>(ev, n, out, ws, cap);
  }
}